// JDCPlusUnet_74096775790994
// MI455X (gfx1250) — compile-verified
//
#include <hip/hip_runtime.h>
#include <hip/hip_bf16.h>
#include <math.h>

// ---------------------------------------------------------------------------
// JDCPlusUnet harmonic-structure kernel for gfx1250 (MI455X, wave32, WMMA)
//
// Outputs (concatenated in d_out, float32):
//   [0 .. 2*256*2049)          harmonic_structure  [B=2,1,T=256,F=2049]
//   [2*256*2049 .. +512)       pitch hz            [B=2,T=256]
// ---------------------------------------------------------------------------

#define N_FFT      4096
#define N_FREQ     2049          // N_FFT/2 + 1
#define N_HARM     301
#define N_HARM_PAD 304           // 76 steps of 4
#define N_STEPS    76
#define N_CLS      722
#define HALF_SR    22050.0f
#define DF         (44100.0f / 4096.0f)   // exactly representable in f32

typedef __attribute__((ext_vector_type(2))) float v2f;
typedef __attribute__((ext_vector_type(8))) float v8f;

// Low 32 bits of a flat shared-aperture pointer are the LDS byte address
// (ISA 10.2).
__device__ __forceinline__ unsigned lds_addr_of(const void* p) {
    return (unsigned)(uintptr_t)p;
}

// ---------------------------------------------------------------------------
// Kernel 1: per-(b,t) argmax over 722 classes -> pitch in Hz.
// One wave32 per (b,t); 8 waves per block.
// ---------------------------------------------------------------------------
__global__ void pitch_hz_kernel(const float* __restrict__ pc,
                                float* __restrict__ hz_out,
                                int n_bt) {
    const int wave = threadIdx.x >> 5;
    const int lane = threadIdx.x & 31;
    const int bt   = blockIdx.x * (blockDim.x >> 5) + wave;
    if (bt >= n_bt) return;                       // wave-uniform

    const float* row = pc + (size_t)bt * N_CLS;
    __builtin_prefetch(row, 0, 0);                // global_prefetch_b8 (gfx1250)

    float bv = -3.402823466e38f;
    int   bi = 0;
    for (int c = lane; c < N_CLS; c += 32) {      // strictly increasing idx per lane
        float v = row[c];
        if (v > bv) { bv = v; bi = c; }           // strict > keeps first max
    }
    // wave32 reduction, tie-break = smallest index (first occurrence)
    #pragma unroll
    for (int off = 16; off > 0; off >>= 1) {
        float ov = __shfl_xor(bv, off, 32);
        int   oi = __shfl_xor(bi, off, 32);
        if (ov > bv || (ov == bv && oi < bi)) { bv = ov; bi = oi; }
    }
    if (lane == 0) {
        float midi = 38.0f + ((float)bi - 1.0f) / 16.0f;
        float hz   = exp2f((midi - 69.0f) / 12.0f) * 440.0f;
        hz_out[bt] = (bi == 0) ? 0.0f : hz;
    }
}

// ---------------------------------------------------------------------------
// Kernel 2: harmonic structure.
// One block per (b,t); 8 waves/block. Each wave owns 16-wide freq tiles and
// walks the 301 harmonics 4 at a time, forming a 4x16 tile of gaussian
// values as a WMMA B-matrix. A = ones(16x4) so
//   D = A*B + C  ==  per-column sum of the 4 gaussians, accumulated in f32.
// Row M=0 of the accumulator (VGPR0, lanes 0-15) is the result for 16 bins.
//
// w/sigma tables are staged into LDS with the gfx1250 async-to-LDS path
// (global_load_async_to_lds_b32, GVS mode) and drained with s_wait_asynccnt.
//
// B-matrix layout (ISA 7.12.2, 32-bit, 4x16 = 2 VGPRs):
//   lanes 0-15 : V0 -> K=0, V1 -> K=1      (N = lane)
//   lanes 16-31: V0 -> K=2, V1 -> K=3      (N = lane-16)
// ---------------------------------------------------------------------------
__global__ void harmonic_kernel(const float* __restrict__ w_h,
                                const float* __restrict__ s_h,
                                const float* __restrict__ hz_arr,
                                float* __restrict__ out) {
    __shared__ float lw[N_HARM_PAD];     // w_n    (pad slots 301..303 = 0)
    __shared__ float ls[N_HARM_PAD];     // s_n    (raw sigma; squared in-loop)

    const int tid = threadIdx.x;

    // --- async stage: 301 w's + 301 sigmas, global -> LDS, no VGPR bounce ---
    {
        const unsigned lw_base = lds_addr_of(&lw[0]);
        const unsigned ls_base = lds_addr_of(&ls[0]);
        for (int i = tid; i < N_HARM; i += (int)blockDim.x) {
            const unsigned voff  = 4u * (unsigned)i;   // byte offset into table
            const unsigned lwa   = lw_base + voff;     // LDS dest byte address
            const unsigned lsa   = ls_base + voff;
            asm volatile("global_load_async_to_lds_b32 %0, %1, %2"
                         :: "v"(lwa), "v"(voff), "s"(w_h) : "memory");
            asm volatile("global_load_async_to_lds_b32 %0, %1, %2"
                         :: "v"(lsa), "v"(voff), "s"(s_h) : "memory");
        }
        if (tid >= N_HARM && tid < N_HARM_PAD) {       // zero the 3 pad slots
            lw[tid] = 0.0f;
            ls[tid] = 0.0f;
        }
        asm volatile("s_wait_asynccnt 0x0" ::: "memory");
    }
    __syncthreads();

    const int   bt    = blockIdx.x;
    const float hz    = hz_arr[bt];
    const bool  valid = (hz > 0.0f) && (hz <= HALF_SR);

    const int lane  = tid & 31;
    const int wave  = tid >> 5;
    const int col   = lane & 15;
    const int half  = lane >> 4;         // 0: K=0/1, 1: K=2/3
    const int kA    = half * 2;          // 0-based K of this lane's V0 slot
    const int nwave = (int)(blockDim.x >> 5);

    float* __restrict__ orow = out + (size_t)bt * N_FREQ;

    v2f a; a[0] = 1.0f; a[1] = 1.0f;     // ones(16x4): every D row = colsum(B)+C

    // 2049 bins -> 129 tiles of 16 (last tile: only column 0 valid)
    for (int tile = wave; tile <= (N_FREQ / 16); tile += nwave) {
        const int   f    = tile * 16 + col;
        const float freq = (float)f * DF;

        v8f c = {};                       // f32 accumulator (16x16 C/D)
        for (int step = 0; step < N_STEPS; ++step) {
            const int h0 = step * 4 + kA;            // 0-based harmonic index
            const int h1 = h0 + 1;

            float m0 = hz * (float)(h0 + 1);
            float d0 = freq - m0;
            float s0 = ls[h0];
            float e0 = d0 * d0 * (s0 * s0 * 0.5f);
            float g0 = (valid && (m0 <= HALF_SR)) ? lw[h0] * expf(-e0) : 0.0f;

            float m1 = hz * (float)(h1 + 1);
            float d1 = freq - m1;
            float s1 = ls[h1];
            float e1 = d1 * d1 * (s1 * s1 * 0.5f);
            float g1 = (valid && (m1 <= HALF_SR)) ? lw[h1] * expf(-e1) : 0.0f;

            v2f b; b[0] = g0; b[1] = g1;
            // D = A(16x4 ones) * B(4x16 gaussians) + C : exact f32 col-sums
            c = __builtin_amdgcn_wmma_f32_16x16x4_f32(
                    /*neg_a=*/false, a, /*neg_b=*/false, b,
                    /*c_mod=*/(short)0, c, /*reuse_a=*/false, /*reuse_b=*/false);
        }
        // Row M=0 lives in c[0] on lanes 0-15 (N = lane)
        if (half == 0 && f < N_FREQ) orow[f] = c[0];
    }
}

// ---------------------------------------------------------------------------
extern "C" void kernel_launch(void* const* d_in, const int* in_sizes, int n_in,
                              void* d_out, int out_size, void* d_ws, size_t ws_size,
                              hipStream_t stream) {
    const float* pc  = (const float*)d_in[0];   // [2,256,722]
    const float* w_h = (const float*)d_in[1];   // [301]
    const float* s_h = (const float*)d_in[2];   // [301]

    float* out = (float*)d_out;
    const int n_bt = in_sizes[0] / N_CLS;       // 512
    const int HS   = n_bt * N_FREQ;             // 1,049,088
    float* hz_out  = out + HS;                  // hz tail of d_out

    pitch_hz_kernel<<<(n_bt + 7) / 8, 256, 0, stream>>>(pc, hz_out, n_bt);
    harmonic_kernel<<<n_bt, 256, 0, stream>>>(w_h, s_h, hz_out, out);

    (void)d_ws; (void)ws_size; (void)n_in; (void)out_size;
}